// TransformerBlock_1391569404451
// MI455X (gfx1250) — compile-verified
//
#include <hip/hip_runtime.h>
#include <hip/hip_bf16.h>

// ---------------------------------------------------------------------------
// TransformerBlock for MI455X (gfx1250), fp32 end-to-end via
// V_WMMA_F32_16X16X4_F32 (wave32 WMMA), flash-attention with 4-way split-K
// per block and (when available) GLOBAL_LOAD_ASYNC_TO_LDS staging.
//
// Shapes: b=4, n=512, m=8192, h=8, dh=16, dim=128, j=8704.
// d_out = out[2048*128] ++ mem_kv_new[4*8*8704*32]   (floats)
// Workspace (floats): xn(256K) q(256K) kv(512K) hv(256K) hn(256K) = 6 MB
// ---------------------------------------------------------------------------

typedef __attribute__((ext_vector_type(2))) float v2f;
typedef __attribute__((ext_vector_type(8))) float v8f;
typedef int v4i __attribute__((vector_size(16)));

#define LN10000 9.210340371976184f   // ln(10000)

#if __has_builtin(__builtin_amdgcn_global_load_async_to_lds_b128) && \
    __has_builtin(__builtin_amdgcn_s_wait_asynccnt)
#define HAVE_ASYNC_LDS 1
// builtin expects (v4i addrspace(1)*, v4i addrspace(3)*, int offset, int cpol)
#define AS1_V4I(p) ((__attribute__((address_space(1))) v4i*)(p))
#define AS3_V4I(p) ((__attribute__((address_space(3))) v4i*)(p))
#else
#define HAVE_ASYNC_LDS 0
#endif

// D = A(16x4) * B(4x16) + C  -- fp32 WMMA, wave32.
__device__ __forceinline__ v8f wmma4(v2f a, v2f b, v8f c) {
  return __builtin_amdgcn_wmma_f32_16x16x4_f32(
      /*neg_a=*/false, a, /*neg_b=*/false, b,
      /*c_mod=*/(short)0, c, /*reuse_a=*/false, /*reuse_b=*/false);
}

// ---------------------------------------------------------------------------
// LayerNorm over rows of 128: one wave per row, 8 rows per 256-thread block.
// ---------------------------------------------------------------------------
__global__ __launch_bounds__(256) void ln_kernel(
    const float* __restrict__ x, const float* __restrict__ g,
    const float* __restrict__ b, float* __restrict__ y, int nrows)
{
  const int row  = blockIdx.x * 8 + (threadIdx.x >> 5);
  const int lane = threadIdx.x & 31;
  if (row >= nrows) return;

  float4 v = *(const float4*)(x + (size_t)row * 128 + lane * 4);
  float s = v.x + v.y + v.z + v.w;
  #pragma unroll
  for (int m = 16; m >= 1; m >>= 1) s += __shfl_xor(s, m, 32);
  const float mu = s * (1.0f / 128.0f);

  const float d0 = v.x - mu, d1 = v.y - mu, d2 = v.z - mu, d3 = v.w - mu;
  float s2 = d0 * d0 + d1 * d1 + d2 * d2 + d3 * d3;
  #pragma unroll
  for (int m = 16; m >= 1; m >>= 1) s2 += __shfl_xor(s2, m, 32);
  const float rstd = rsqrtf(s2 * (1.0f / 128.0f) + 1e-5f);

  const float4 gg = *(const float4*)(g + lane * 4);
  const float4 bb = *(const float4*)(b + lane * 4);
  float4 o;
  o.x = d0 * rstd * gg.x + bb.x;
  o.y = d1 * rstd * gg.y + bb.y;
  o.z = d2 * rstd * gg.z + bb.z;
  o.w = d3 * rstd * gg.w + bb.w;
  *(float4*)(y + (size_t)row * 128 + lane * 4) = o;
}

// ---------------------------------------------------------------------------
// C[M,N] = A[M,128] @ W[N,128]^T (+ bias[N]) (+ residual[M,N]).
// 8 waves/block; wave w computes the 16x16 tile (blockIdx.x, blockIdx.y*8+w).
// A tile staged in LDS; K=128 fully unrolled -> 32 v_wmma_f32_16x16x4_f32.
// ---------------------------------------------------------------------------
__global__ __launch_bounds__(256) void gemm16_kernel(
    const float* __restrict__ A, const float* __restrict__ W,
    float* __restrict__ C, int M, int N,
    const float* __restrict__ bias, const float* __restrict__ residual)
{
  __shared__ float sA[16][132];   // 132: keeps float4 LDS stores 16B aligned
  const int tid  = threadIdx.x;
  const int lane = tid & 31;
  const int wv   = tid >> 5;
  const int half = lane >> 4;     // K sub-select within fragment
  const int l16  = lane & 15;     // M (for A) / N (for B) index
  const int m0 = blockIdx.x * 16;
  const int n0 = (blockIdx.y * 8 + wv) * 16;

  for (int i = tid; i < 512; i += 256) {
    const int r  = i >> 5;
    const int c4 = (i & 31) << 2;
    *(float4*)&sA[r][c4] = *(const float4*)(A + (size_t)(m0 + r) * 128 + c4);
  }
  __syncthreads();

  v8f acc = {};
  const float* wrow = W + (size_t)(n0 + l16) * 128;   // B[k][n] = W[n][k]
  #pragma unroll
  for (int k = 0; k < 128; k += 4) {
    const v2f a = *(const v2f*)&sA[l16][k + 2 * half];
    const v2f b = *(const v2f*)(wrow + k + 2 * half);
    acc = wmma4(a, b, acc);
  }

  #pragma unroll
  for (int r = 0; r < 8; r++) {
    const int row = m0 + r + 8 * half;
    const int col = n0 + l16;
    float v = acc[r];
    if (bias)     v += bias[col];
    if (residual) v += residual[(size_t)row * N + col];
    C[(size_t)row * N + col] = v;
  }
}

// ---------------------------------------------------------------------------
// Flash attention, 4-way split-K: block = 128 threads (4 waves) per
// (b, h, 16-query tile) -> 1024 blocks, 4096 waves. Wave w processes key
// chunks ck = w, w+4, ... (136 each) with private online-softmax state;
// partials merged through LDS. Raw (k|v) rows staged into per-wave LDS tiles
// via GLOBAL_LOAD_ASYNC_TO_LDS_B128 when available.
//   q:     [2048][128]   kvnew: [2048][256]   memkv: [4][8][8192][32]
//   hv:    [2048][128] = attn out + residual x
// ---------------------------------------------------------------------------
__global__ __launch_bounds__(128) void attn_kernel(
    const float* __restrict__ q, const float* __restrict__ kvnew,
    const float* __restrict__ memkv, const float* __restrict__ x,
    float* __restrict__ hv)
{
  __shared__ float sQ[16][18];
  __shared__ __align__(16) float sRaw[4][16][32];   // per-wave raw (k|v) tile
  __shared__ float sK[4][16][18];                   // per-wave rotated K
  __shared__ float sP[4][16][18];                   // per-wave P transpose
  __shared__ float sOc[4][16][16];                  // partial O
  __shared__ float sStat[4][16][2];                 // partial (m, l)

  const int tid  = threadIdx.x;
  const int lane = tid & 31;
  const int wv   = tid >> 5;
  const int half = lane >> 4;
  const int l16  = lane & 15;
  const int qt = blockIdx.x & 31;
  const int hh = (blockIdx.x >> 5) & 7;
  const int bb = blockIdx.x >> 8;

  // rotary inv_freq depends only on (lane & 7) in all staging loops below
  const float invf = __expf(-(float)(l16 & 7) * (2.0f / 16.0f) * LN10000);

  // ---- stage Q tile with rotary + 1/sqrt(dh): 2 elements per thread ----
  for (int e = tid; e < 256; e += 128) {
    const int r = e >> 4, c = e & 15;          // c == l16
    const int tok = bb * 512 + qt * 16 + r;
    const float* base = q + (size_t)tok * 128 + hh * 16;
    const float qd = base[c];
    const float qp = base[c ^ 8];
    const float fr = (float)(8192 + qt * 16 + r) * invf;
    float sn, cs; __sincosf(fr, &sn, &cs);
    const float rot = (c < 8) ? -qp : qp;
    sQ[r][c] = (qd * cs + rot * sn) * 0.25f;   // 16^-0.5
  }
  __syncthreads();

  v2f qA[4];
  #pragma unroll
  for (int kk = 0; kk < 4; kk++)
    qA[kk] = *(const v2f*)&sQ[l16][4 * kk + 2 * half];

  v8f O = {};
  float rmax[8], rsum[8];
  #pragma unroll
  for (int r = 0; r < 8; r++) { rmax[r] = -3.0e38f; rsum[r] = 0.0f; }

  float* raw = &sRaw[wv][0][0];

  for (int it = 0; it < 136; ++it) {
    const int ck = wv + 4 * it;                // 0..543, chunk-aligned old/new
    const int j0 = ck * 16;
    const bool oldc = (j0 < 8192);

    // ---- stage raw [16 rows x 32 floats] -> per-wave LDS (4 x b128/lane) ----
    #pragma unroll
    for (int i = 0; i < 4; ++i) {
      const int f  = lane + 32 * i;            // 0..127 float4 slots
      const int r  = f >> 3;
      const int q4 = (f & 7) * 4;
      const float* gsrc = oldc
        ? memkv + ((size_t)(bb * 8 + hh) * 8192 + (j0 + r)) * 32 + q4
        : kvnew + (size_t)(bb * 512 + (j0 - 8192 + r)) * 256 + hh * 32 + q4;
      float* ldst = raw + r * 32 + q4;
#if HAVE_ASYNC_LDS
      __builtin_amdgcn_global_load_async_to_lds_b128(
          AS1_V4I(gsrc), AS3_V4I(ldst), /*offset=*/0, /*cpol=*/0);
#else
      *(float4*)ldst = *(const float4*)gsrc;
#endif
    }
#if HAVE_ASYNC_LDS
    __builtin_amdgcn_s_wait_asynccnt(0);
#endif
    __syncthreads();

    // ---- rotary pass on K (8 elems/lane, c == l16 fixed) ----
    #pragma unroll
    for (int i = 0; i < 8; ++i) {
      const int r = half + 2 * i;
      const int c = l16;
      const int jj = j0 + r;
      const float kd = raw[r * 32 + c];
      const float kp = raw[r * 32 + (c ^ 8)];
      float sn, cs; __sincosf((float)jj * invf, &sn, &cs);
      const float rot = (c < 8) ? -kp : kp;
      sK[wv][r][c] = kd * cs + rot * sn;
    }
    __syncthreads();

    // ---- S = Q @ K^T ----
    v8f s = {};
    #pragma unroll
    for (int kk = 0; kk < 4; kk++) {
      const v2f b = *(const v2f*)&sK[wv][l16][4 * kk + 2 * half];
      s = wmma4(qA[kk], b, s);
    }

    // ---- online softmax (C-layout row = r + 8*half, col = l16) ----
    #pragma unroll
    for (int r = 0; r < 8; r++) {
      const float sv = s[r];
      float mx = sv;
      mx = fmaxf(mx, __shfl_xor(mx, 1, 32));
      mx = fmaxf(mx, __shfl_xor(mx, 2, 32));
      mx = fmaxf(mx, __shfl_xor(mx, 4, 32));
      mx = fmaxf(mx, __shfl_xor(mx, 8, 32));
      const float mn    = fmaxf(rmax[r], mx);
      const float alpha = __expf(rmax[r] - mn);
      const float p     = __expf(sv - mn);
      float ps = p;
      ps += __shfl_xor(ps, 1, 32);
      ps += __shfl_xor(ps, 2, 32);
      ps += __shfl_xor(ps, 4, 32);
      ps += __shfl_xor(ps, 8, 32);
      rsum[r] = rsum[r] * alpha + ps;
      rmax[r] = mn;
      O[r] *= alpha;
      sP[wv][r + 8 * half][l16] = p;
    }
    __syncthreads();

    // ---- O += P @ V  (V read straight from raw tile, cols 16..31) ----
    #pragma unroll
    for (int kk = 0; kk < 4; kk++) {
      const v2f pa = *(const v2f*)&sP[wv][l16][4 * kk + 2 * half];
      const int kr = 4 * kk + 2 * half;
      v2f vb;
      vb.x = raw[kr * 32 + 16 + l16];
      vb.y = raw[(kr + 1) * 32 + 16 + l16];
      O = wmma4(pa, vb, O);
    }
    __syncthreads();
  }

  // ---- publish per-wave partials ----
  #pragma unroll
  for (int r = 0; r < 8; r++)
    sOc[wv][r + 8 * half][l16] = O[r];
  if (l16 == 0) {
    #pragma unroll
    for (int r = 0; r < 8; r++) {
      sStat[wv][r + 8 * half][0] = rmax[r];
      sStat[wv][r + 8 * half][1] = rsum[r];
    }
  }
  __syncthreads();

  // ---- wave 0 merges the 4 partials, adds residual, writes hv ----
  if (wv == 0) {
    for (int e = lane; e < 256; e += 32) {
      const int row = e >> 4, col = e & 15;
      float M = sStat[0][row][0];
      M = fmaxf(M, sStat[1][row][0]);
      M = fmaxf(M, sStat[2][row][0]);
      M = fmaxf(M, sStat[3][row][0]);
      float den = 0.0f, num = 0.0f;
      #pragma unroll
      for (int w = 0; w < 4; w++) {
        const float sc = __expf(sStat[w][row][0] - M);
        den += sStat[w][row][1] * sc;
        num += sOc[w][row][col] * sc;
      }
      const int tok = bb * 512 + qt * 16 + row;
      const int cl  = hh * 16 + col;
      hv[(size_t)tok * 128 + cl] = num / den + x[(size_t)tok * 128 + cl];
    }
  }
}

// ---------------------------------------------------------------------------
// mem_kv_out[4][8][8704][32] = concat(memkv old, kv new) ; float4 per thread.
// ---------------------------------------------------------------------------
__global__ __launch_bounds__(256) void memkv_cat_kernel(
    const float4* __restrict__ memkv, const float* __restrict__ kvnew,
    float4* __restrict__ out)
{
  const size_t idx = (size_t)blockIdx.x * 256 + threadIdx.x;  // one float4
  const size_t total = (size_t)4 * 8 * 8704 * 8;
  if (idx >= total) return;
  const int    f4  = (int)(idx & 7);
  const size_t row = idx >> 3;
  const int jj = (int)(row % 8704);
  const int bh = (int)(row / 8704);
  float4 v;
  if (jj < 8192) {
    v = memkv[((size_t)bh * 8192 + jj) * 8 + f4];
  } else {
    const int t = jj - 8192, bb = bh >> 3, hh = bh & 7;
    v = *(const float4*)(kvnew + ((size_t)(bb * 512 + t)) * 256 + hh * 32 + f4 * 4);
  }
  out[idx] = v;
}

// ---------------------------------------------------------------------------
extern "C" void kernel_launch(void* const* d_in, const int* in_sizes, int n_in,
                              void* d_out, int out_size, void* d_ws, size_t ws_size,
                              hipStream_t stream)
{
  (void)in_sizes; (void)n_in; (void)out_size; (void)ws_size;

  const float* x     = (const float*)d_in[0];
  const float* memkv = (const float*)d_in[1];
  const float* w_q   = (const float*)d_in[2];
  const float* w_kv  = (const float*)d_in[3];
  const float* w_out = (const float*)d_in[4];
  const float* b_out = (const float*)d_in[5];
  const float* g1    = (const float*)d_in[6];
  const float* b1    = (const float*)d_in[7];
  const float* g2    = (const float*)d_in[8];
  const float* b2    = (const float*)d_in[9];

  float* out   = (float*)d_out;            // [2048][128]
  float* mkout = out + 2048 * 128;         // [4][8][8704][32]

  float* ws = (float*)d_ws;                // needs 6 MB of fp32 scratch
  float* xn = ws;                          // [2048][128] LN1(x)
  float* qb = ws + 262144;                 // [2048][128] q proj
  float* kv = ws + 524288;                 // [2048][256] kv proj
  float* hb = ws + 1048576;                // [2048][128] attn out + x
  float* hn = ws + 1310720;                // [2048][128] LN2(hv)

  ln_kernel<<<256, 256, 0, stream>>>(x, g1, b1, xn, 2048);
  gemm16_kernel<<<dim3(128, 1), 256, 0, stream>>>(xn, w_q, qb, 2048, 128, nullptr, nullptr);
  gemm16_kernel<<<dim3(128, 2), 256, 0, stream>>>(xn, w_kv, kv, 2048, 256, nullptr, nullptr);
  attn_kernel<<<1024, 128, 0, stream>>>(qb, kv, memkv, x, hb);
  ln_kernel<<<256, 256, 0, stream>>>(hb, g2, b2, hn, 2048);
  gemm16_kernel<<<dim3(128, 1), 256, 0, stream>>>(hn, w_out, out, 2048, 128, b_out, hb);
  memkv_cat_kernel<<<8704, 256, 0, stream>>>((const float4*)memkv, kv, (float4*)mkout);
}